// TrajectoryInference_15006615733716
// MI455X (gfx1250) — compile-verified
//
#include <hip/hip_runtime.h>
#include <math.h>

// ---------------------------------------------------------------------------
// Trajectory inference pipeline for MI455X (gfx1250, wave32).
// All GEMMs use V_WMMA_F32_16X16X4_F32 (full fp32 precision — needed for the
// subspace/eigen iteration; total pipeline is only ~15 GFLOP, far below the
// fp32 matrix-op ceiling, and the ~90 MB working set fits in the 192 MB L2,
// so the design goal is a clean WMMA inner loop + device-resident data).
// ---------------------------------------------------------------------------

typedef __attribute__((ext_vector_type(2))) float v2f;
typedef __attribute__((ext_vector_type(8))) float v8f;

#define ROWS 4096      // cells (n)
#define COLS 1000      // genes
#define NPC  50        // principal components kept
#define NPCP 64        // padded PC count (multiple of 16 for WMMA tiles)
#define KNN  15
#define SUBSPACE_ITERS 30
#define DIFF_ITERS 100

// ---------------------------------------------------------------------------
// fp32 WMMA helpers
// ---------------------------------------------------------------------------
__device__ __forceinline__ v8f wmma4(const v2f a, const v2f b, const v8f c)
{
    // (neg_a, A, neg_b, B, c_mod, C, reuse_a, reuse_b)
    return __builtin_amdgcn_wmma_f32_16x16x4_f32(
        false, a, false, b, (short)0, c, false, false);
}

// C/D layout: VGPR r -> row = i0 + 8*(lane>>4) + r, col = j0 + (lane&15)
__device__ __forceinline__ void store_tile(float* __restrict__ C, const v8f acc,
                                           int i0, int j0, int M, int N,
                                           int ldc, float alpha, int lane)
{
    const int col = j0 + (lane & 15);
    const int rb  = i0 + ((lane >> 4) << 3);
    if (col < N) {
#pragma unroll
        for (int r = 0; r < 8; ++r) {
            const int row = rb + r;
            if (row < M) C[(size_t)row * ldc + col] = alpha * acc[r];
        }
    }
}

// ---------------------------------------------------------------------------
// Shared 2x2-register-blocked K loop.  MASKED=false is the interior fast
// path: pure loads + WMMA, no per-element mask multiplies.
// ---------------------------------------------------------------------------
template <bool TA, bool TB, bool MASKED>
__device__ __forceinline__ void gemm_kloop(
    const float* __restrict__ A, const float* __restrict__ B,
    int K, int lda, int ldb, int kb,
    int aS0, int aS1, int bS0, int bS1,
    float mA0, float mA1, float mB0, float mB1,
    v8f& acc00, v8f& acc01, v8f& acc10, v8f& acc11)
{
#pragma unroll 2
    for (int k0 = 0; k0 < K; k0 += 4) {
        const int kLo = k0 + kb;
        const int kHi = kLo + 1;
        v2f a0, a1, b0, b1;
        if (!TA) {  // contiguous along K -> float2 (b64) loads
            const float2 t0 = *(const float2*)(A + (size_t)aS0 * lda + kLo);
            const float2 t1 = *(const float2*)(A + (size_t)aS1 * lda + kLo);
            a0.x = t0.x; a0.y = t0.y;
            a1.x = t1.x; a1.y = t1.y;
        } else {    // K-major stride -> coalesced across lanes 0..15
            a0.x = A[(size_t)kLo * lda + aS0];
            a0.y = A[(size_t)kHi * lda + aS0];
            a1.x = A[(size_t)kLo * lda + aS1];
            a1.y = A[(size_t)kHi * lda + aS1];
        }
        if (!TB) {  // coalesced across lanes 0..15
            b0.x = B[(size_t)kLo * ldb + bS0];
            b0.y = B[(size_t)kHi * ldb + bS0];
            b1.x = B[(size_t)kLo * ldb + bS1];
            b1.y = B[(size_t)kHi * ldb + bS1];
        } else {    // contiguous along K -> float2 (b64) loads
            const float2 t0 = *(const float2*)(B + (size_t)bS0 * ldb + kLo);
            const float2 t1 = *(const float2*)(B + (size_t)bS1 * ldb + kLo);
            b0.x = t0.x; b0.y = t0.y;
            b1.x = t1.x; b1.y = t1.y;
        }
        if (MASKED) {   // zero out-of-range rows/cols (clamped addresses)
            a0.x *= mA0; a0.y *= mA0; a1.x *= mA1; a1.y *= mA1;
            b0.x *= mB0; b0.y *= mB0; b1.x *= mB1; b1.y *= mB1;
        }
        acc00 = wmma4(a0, b0, acc00);
        acc01 = wmma4(a0, b1, acc01);
        acc10 = wmma4(a1, b0, acc10);
        acc11 = wmma4(a1, b1, acc11);
    }
}

// ---------------------------------------------------------------------------
// Generic fp32 WMMA GEMM: C = alpha * opA(A) * opB(B)
// opA(A) is M x K, opB(B) is K x N.  TA: A[m][k] = A[k*lda + m]
//                                    TB: B[k][n] = B[n*ldb + k]
// One wave computes a 32x32 C macro-tile as 2x2 WMMA tiles (register
// blocking: each A/B fragment feeds two WMMAs).  K must be a multiple of 4.
// Interior tiles take a mask-free fast path; edge tiles use clamped
// addresses + zero multipliers so EXEC stays all-ones around the WMMAs
// (ISA requirement) with no branches inside the K loop.
// ---------------------------------------------------------------------------
template <bool TA, bool TB>
__global__ __launch_bounds__(32)
void wmma_gemm_f32(const float* __restrict__ A, const float* __restrict__ B,
                   float* __restrict__ C, int M, int N, int K,
                   int lda, int ldb, int ldc, float alpha)
{
    const int j0   = blockIdx.x * 32;
    const int i0   = blockIdx.y * 32;
    const int lane = threadIdx.x;
    const int mn   = lane & 15;
    const int kb   = (lane >> 4) << 1;      // 0 or 2

    const int am0 = i0 + mn,      am1 = i0 + 16 + mn;   // A fragment M indices
    const int bn0 = j0 + mn,      bn1 = j0 + 16 + mn;   // B fragment N indices

    v8f acc00 = {0.f,0.f,0.f,0.f,0.f,0.f,0.f,0.f};
    v8f acc01 = acc00, acc10 = acc00, acc11 = acc00;

    if (i0 + 32 <= M && j0 + 32 <= N) {
        // Interior: no masking at all.
        gemm_kloop<TA, TB, false>(A, B, K, lda, ldb, kb,
                                  am0, am1, bn0, bn1,
                                  1.f, 1.f, 1.f, 1.f,
                                  acc00, acc01, acc10, acc11);
    } else {
        const float mA0 = (am0 < M) ? 1.f : 0.f, mA1 = (am1 < M) ? 1.f : 0.f;
        const float mB0 = (bn0 < N) ? 1.f : 0.f, mB1 = (bn1 < N) ? 1.f : 0.f;
        const int aS0 = (am0 < M) ? am0 : 0, aS1 = (am1 < M) ? am1 : 0;
        const int bS0 = (bn0 < N) ? bn0 : 0, bS1 = (bn1 < N) ? bn1 : 0;
        gemm_kloop<TA, TB, true>(A, B, K, lda, ldb, kb,
                                 aS0, aS1, bS0, bS1,
                                 mA0, mA1, mB0, mB1,
                                 acc00, acc01, acc10, acc11);
    }

    store_tile(C, acc00, i0,      j0,      M, N, ldc, alpha, lane);
    store_tile(C, acc01, i0,      j0 + 16, M, N, ldc, alpha, lane);
    store_tile(C, acc10, i0 + 16, j0,      M, N, ldc, alpha, lane);
    store_tile(C, acc11, i0 + 16, j0 + 16, M, N, ldc, alpha, lane);
}

// ---------------------------------------------------------------------------
__global__ void colmean_kernel(const float* __restrict__ x,
                               float* __restrict__ mean, int rows, int cols)
{
    const int c = blockIdx.x * blockDim.x + threadIdx.x;
    if (c >= cols) return;
    float s = 0.f;
    for (int r = 0; r < rows; ++r) s += x[(size_t)r * cols + c];
    mean[c] = s / (float)rows;
}

__global__ void center_kernel(const float* __restrict__ x,
                              const float* __restrict__ mean,
                              float* __restrict__ xc, int total, int cols)
{
    const int i = blockIdx.x * blockDim.x + threadIdx.x;
    if (i >= total) return;
    xc[i] = x[i] - mean[i % cols];
}

// Deterministic pseudo-random init for the subspace iteration basis.
__global__ void init_v_kernel(float* __restrict__ V, int total)
{
    const int i = blockIdx.x * blockDim.x + threadIdx.x;
    if (i >= total) return;
    unsigned h = (unsigned)i * 2654435761u;
    h ^= h >> 13; h *= 0x9E3779B1u; h ^= h >> 16;
    V[i] = (float)(h & 0xFFFFu) * (1.f / 65536.f) - 0.5f;
}

// Zero padded eigenvector columns [NPC, NPCP) so Gram/d2 only sees 50 PCs.
__global__ void zero_tail_cols_kernel(float* __restrict__ V, int rows)
{
    const int i = blockIdx.x * blockDim.x + threadIdx.x;
    const int tail = NPCP - NPC;
    if (i >= rows * tail) return;
    const int r = i / tail, c = NPC + (i % tail);
    V[r * NPCP + c] = 0.f;
}

// ---------------------------------------------------------------------------
// Modified Gram-Schmidt QR of V (rows x cols, row-major), single block.
// ---------------------------------------------------------------------------
__global__ __launch_bounds__(1024)
void mgs_kernel(float* __restrict__ V, int rows, int cols)
{
    __shared__ float red[1024];
    const int tid = threadIdx.x;
    const int bs  = blockDim.x;
    for (int j = 0; j < cols; ++j) {
        float s = 0.f;
        for (int r = tid; r < rows; r += bs) {
            const float v = V[r * cols + j]; s += v * v;
        }
        red[tid] = s; __syncthreads();
        for (int off = bs >> 1; off > 0; off >>= 1) {
            if (tid < off) red[tid] += red[tid + off];
            __syncthreads();
        }
        const float inv = rsqrtf(red[0] + 1e-30f);
        __syncthreads();
        for (int r = tid; r < rows; r += bs) V[r * cols + j] *= inv;
        __syncthreads();
        for (int i = j + 1; i < cols; ++i) {
            float d = 0.f;
            for (int r = tid; r < rows; r += bs)
                d += V[r * cols + j] * V[r * cols + i];
            red[tid] = d; __syncthreads();
            for (int off = bs >> 1; off > 0; off >>= 1) {
                if (tid < off) red[tid] += red[tid + off];
                __syncthreads();
            }
            const float dj = red[0];
            __syncthreads();
            for (int r = tid; r < rows; r += bs)
                V[r * cols + i] -= dj * V[r * cols + j];
            __syncthreads();
        }
    }
}

// ---------------------------------------------------------------------------
__global__ void sqnorm_kernel(const float* __restrict__ P, float* __restrict__ sq)
{
    const int r = blockIdx.x * blockDim.x + threadIdx.x;
    if (r >= ROWS) return;
    float s = 0.f;
    for (int c = 0; c < NPCP; ++c) { const float v = P[r * NPCP + c]; s += v * v; }
    sq[r] = s;
}

// d2[i][j] = max(sq[i] + sq[j] - 2*G[i][j], 0), in place over G.
__global__ void d2_kernel(float* __restrict__ G, const float* __restrict__ sq, int n)
{
    const int idx = blockIdx.x * blockDim.x + threadIdx.x;
    if (idx >= n * n) return;
    const int i = idx / n, j = idx % n;
    const float d = sq[i] + sq[j] - 2.f * G[idx];
    G[idx] = fmaxf(d, 0.f);
}

__global__ void zero_f32_kernel(float* __restrict__ p, int total)
{
    const int i = blockIdx.x * blockDim.x + threadIdx.x;
    if (i < total) p[i] = 0.f;
}

// ---------------------------------------------------------------------------
// Per-row 15-NN: per-thread insertion-sorted top-15 over 32 strided columns,
// LDS merge, serial select of the 15 global minima.  Writes normalized
// Gaussian weights: dense transition row (scale 1/(2*sigma^2)=0.5) and sparse
// diffusion weights (scale 1.0) + indices for the power iteration.
// ---------------------------------------------------------------------------
__global__ __launch_bounds__(128)
void topk_kernel(const float* __restrict__ d2, int n,
                 int* __restrict__ kidx, float* __restrict__ kwd,
                 float* __restrict__ trans)
{
    const int row = blockIdx.x;
    const int tid = threadIdx.x;
    const float* drow = d2 + (size_t)row * n;

    float v[KNN]; int id[KNN];
#pragma unroll
    for (int k = 0; k < KNN; ++k) { v[k] = 3.4e38f; id[k] = -1; }
    for (int c = tid; c < n; c += 128) {
        const float x = drow[c];
        if (x < v[KNN - 1]) {
            int p = KNN - 1;
            while (p > 0 && v[p - 1] > x) {
                v[p] = v[p - 1]; id[p] = id[p - 1]; --p;
            }
            v[p] = x; id[p] = c;
        }
    }
    __shared__ float cv[128 * KNN];
    __shared__ int   ci[128 * KNN];
#pragma unroll
    for (int k = 0; k < KNN; ++k) {
        cv[tid * KNN + k] = v[k];
        ci[tid * KNN + k] = id[k];
    }
    __syncthreads();
    if (tid == 0) {
        float selv[KNN]; int seli[KNN];
        for (int k = 0; k < KNN; ++k) {
            int bp = 0;
            for (int c = 1; c < 128 * KNN; ++c) {
                if (cv[c] < cv[bp] ||
                    (cv[c] == cv[bp] && ci[c] >= 0 && ci[c] < ci[bp])) bp = c;
            }
            selv[k] = cv[bp]; seli[k] = ci[bp]; cv[bp] = 3.4e38f;
        }
        float st = 0.f, sd = 0.f, wt[KNN], wd[KNN];
        for (int k = 0; k < KNN; ++k) {
            wt[k] = __expf(-0.5f * selv[k]);   // transition scale 1/(2*sigma^2)
            wd[k] = __expf(-selv[k]);          // diffusion scale 1.0
            st += wt[k]; sd += wd[k];
        }
        for (int k = 0; k < KNN; ++k) {
            kidx[row * KNN + k] = seli[k];
            kwd[row * KNN + k]  = wd[k] / sd;
            trans[(size_t)row * n + seli[k]] = wt[k] / st;
        }
    }
}

// ---------------------------------------------------------------------------
// 100 power iterations of the sparse (15 nnz/row) diffusion operator with
// max-normalization, entirely inside one block (no per-iteration launches).
// ---------------------------------------------------------------------------
__global__ __launch_bounds__(1024)
void diffuse_kernel(const int* __restrict__ kidx, const float* __restrict__ kwd,
                    float* __restrict__ dbuf, float* __restrict__ pt_out, int n)
{
    __shared__ float red[1024];
    const int tid = threadIdx.x;
    float* cur = dbuf;
    float* nxt = dbuf + n;
    for (int r = tid; r < n; r += 1024) cur[r] = (r == 0) ? 1.f : 0.f;
    __syncthreads();
    for (int it = 0; it < DIFF_ITERS; ++it) {
        float lmax = -3.4e38f;
        for (int r = tid; r < n; r += 1024) {
            float s = 0.f;
#pragma unroll
            for (int k = 0; k < KNN; ++k)
                s += kwd[r * KNN + k] * cur[kidx[r * KNN + k]];
            nxt[r] = s;
            lmax = fmaxf(lmax, s);
        }
        red[tid] = lmax; __syncthreads();
        for (int off = 512; off > 0; off >>= 1) {
            if (tid < off) red[tid] = fmaxf(red[tid], red[tid + off]);
            __syncthreads();
        }
        const float inv = 1.f / red[0];
        __syncthreads();
        for (int r = tid; r < n; r += 1024) nxt[r] *= inv;
        float* t = cur; cur = nxt; nxt = t;
        __syncthreads();
    }
    for (int r = tid; r < n; r += 1024) pt_out[r] = cur[r];
}

// Copy first 50 of the 64 padded PC columns to the output.
__global__ void pca_out_kernel(const float* __restrict__ P, float* __restrict__ out)
{
    const int i = blockIdx.x * blockDim.x + threadIdx.x;
    if (i >= ROWS * NPC) return;
    const int r = i / NPC, c = i % NPC;
    out[i] = P[r * NPCP + c];
}

// ---------------------------------------------------------------------------
extern "C" void kernel_launch(void* const* d_in, const int* in_sizes, int n_in,
                              void* d_out, int out_size, void* d_ws, size_t ws_size,
                              hipStream_t stream)
{
    (void)in_sizes; (void)n_in; (void)out_size; (void)ws_size;
    const float* x = (const float*)d_in[0];
    float* out = (float*)d_out;
    const int n = ROWS;

    // Workspace layout (floats, 256B-aligned blocks). Total ~90 MB.
    float* ws = (float*)d_ws;
    size_t off = 0;
    auto alloc = [&](size_t nf) {
        float* p = ws + off;
        off += (nf + 63) & ~(size_t)63;
        return p;
    };
    float* xc   = alloc((size_t)ROWS * COLS);   // centered expression
    float* mean = alloc(COLS);
    float* cov  = alloc((size_t)COLS * COLS);   // gene covariance
    float* Va   = alloc((size_t)COLS * NPCP);   // subspace basis (ping)
    float* Vb   = alloc((size_t)COLS * NPCP);   // subspace basis (pong)
    float* P    = alloc((size_t)ROWS * NPCP);   // padded PCA scores
    float* sq   = alloc(ROWS);
    float* G    = alloc((size_t)n * n);         // Gram -> d2 in place
    float* kwd  = alloc((size_t)n * KNN);       // diffusion weights
    int*   kidx = (int*)alloc((size_t)n * KNN); // NN indices
    float* dbuf = alloc(2 * (size_t)n);         // power-iteration ping-pong

    // Output layout: pseudotime[4096] | transition[4096*4096] | pca[4096*50]
    float* pt_out    = out;
    float* trans_out = out + n;
    float* pca_out   = out + n + (size_t)n * n;

    // 1. center columns
    colmean_kernel<<<(COLS + 255) / 256, 256, 0, stream>>>(x, mean, ROWS, COLS);
    center_kernel<<<(ROWS * COLS + 255) / 256, 256, 0, stream>>>(
        x, mean, xc, ROWS * COLS, COLS);

    // 2. cov = xc^T xc / (n-1)    [WMMA, M=N=1000, K=4096, A transposed]
    {
        dim3 grid((COLS + 31) / 32, (COLS + 31) / 32);
        wmma_gemm_f32<true, false><<<grid, 32, 0, stream>>>(
            xc, xc, cov, COLS, COLS, ROWS, COLS, COLS, COLS,
            1.f / (float)(ROWS - 1));
    }

    // 3. top-50 subspace via orthogonal iteration (64 padded columns):
    //    repeat: W = cov @ V  [WMMA]  ;  V = MGS-QR(W)
    init_v_kernel<<<(COLS * NPCP + 255) / 256, 256, 0, stream>>>(Va, COLS * NPCP);
    float* curV = Va; float* othV = Vb;
    for (int t = 0; t < SUBSPACE_ITERS; ++t) {
        dim3 grid((NPCP + 31) / 32, (COLS + 31) / 32);
        wmma_gemm_f32<false, false><<<grid, 32, 0, stream>>>(
            cov, curV, othV, COLS, NPCP, COLS, COLS, NPCP, NPCP, 1.f);
        mgs_kernel<<<1, 1024, 0, stream>>>(othV, COLS, NPCP);
        float* tmp = curV; curV = othV; othV = tmp;
    }
    zero_tail_cols_kernel<<<(COLS * (NPCP - NPC) + 255) / 256, 256, 0, stream>>>(
        curV, COLS);

    // 4. P = xc @ V   [WMMA, M=4096, N=64, K=1000] (cols 50..63 are zero)
    {
        dim3 grid((NPCP + 31) / 32, (ROWS + 31) / 32);
        wmma_gemm_f32<false, false><<<grid, 32, 0, stream>>>(
            xc, curV, P, ROWS, NPCP, COLS, COLS, NPCP, NPCP, 1.f);
    }
    pca_out_kernel<<<(ROWS * NPC + 255) / 256, 256, 0, stream>>>(P, pca_out);

    // 5. Gram = P P^T  [WMMA, M=N=4096, K=64, B transposed], then d2 in place
    sqnorm_kernel<<<(ROWS + 255) / 256, 256, 0, stream>>>(P, sq);
    {
        dim3 grid((n + 31) / 32, (n + 31) / 32);
        wmma_gemm_f32<false, true><<<grid, 32, 0, stream>>>(
            P, P, G, n, n, NPCP, NPCP, NPCP, n, 1.f);
    }
    d2_kernel<<<(n * n + 255) / 256, 256, 0, stream>>>(G, sq, n);

    // 6. 15-NN affinities -> dense transition (pre-zeroed) + sparse diffusion
    zero_f32_kernel<<<(n * n + 255) / 256, 256, 0, stream>>>(trans_out, n * n);
    topk_kernel<<<n, 128, 0, stream>>>(G, n, kidx, kwd, trans_out);

    // 7. 100 max-normalized diffusion power iterations -> pseudotime
    diffuse_kernel<<<1, 1024, 0, stream>>>(kidx, kwd, dbuf, pt_out, n);
}